// YOLOV2Loss_11665131176540
// MI455X (gfx1250) — compile-verified
//
#include <hip/hip_runtime.h>
#include <hip/hip_bf16.h>

// ---- problem constants (from reference setup_inputs) ----
#define BB   64          // batch
#define AA   5           // anchors
#define CC   80          // classes
#define GG   32          // grid (GY == GX == 32)
#define NN   50          // gt boxes per image
#define CH   85          // 5 + C
#define NBX  (AA*GG*GG)  // 5120 boxes per image
#define NGT  (BB*NN)     // 3200 gt boxes total
#define MAINBLKS ((NBX/256)*BB)   // 20*64 = 1280
#define GTBLKS   (NGT/8)          // 400 (8 warps of 32 per block)
#define NPART    (MAINBLKS + GTBLKS)

#define L_OBJ   5.0f
#define L_NOOBJ 1.0f
#define L_PRIOR 0.01f
#define L_COORD 1.0f
#define IOU_THR 0.6f
#define SEEN_THR 12800

// ---- workspace layout (bytes) ----
#define WS_BESTBOX 0                    // int[3200]
#define WS_ENC     12800                // float[3200*4]
#define WS_GTX     64000                // float[3200*4]  (gt boxes xyxy)
#define WS_FLAG    115200               // uchar[BB*NBX] = 327680
#define WS_PART    442880               // float[NPART]

typedef __attribute__((ext_vector_type(2))) float v2f;
typedef __attribute__((ext_vector_type(8))) float v8f;

__device__ __forceinline__ float sigmoidf(float x) { return 1.0f / (1.0f + __expf(-x)); }

__device__ __forceinline__ float iou_xyxy(float ax1, float ay1, float ax2, float ay2,
                                          float areaA,
                                          float bx1, float by1, float bx2, float by2,
                                          float areaB) {
    float lx = fmaxf(ax1, bx1), ly = fmaxf(ay1, by1);
    float rx = fminf(ax2, bx2), ry = fminf(ay2, by2);
    float w = fmaxf(rx - lx, 0.0f), h = fmaxf(ry - ly, 0.0f);
    float inter = w * h;
    return inter / (areaA + areaB - inter);
}

// ---------------- kernel 0: zero the matched-box flags ----------------
__global__ void k_zero_flags(unsigned int* flags_u32) {
    int i = blockIdx.x * blockDim.x + threadIdx.x;           // 81920 words
    if (i < (BB * NBX) / 4) flags_u32[i] = 0u;
}

// ---------------- kernel 1: per-GT prep ----------------
// best anchor (wh IoU argmax, first-max like jnp.argmax), best_box scatter,
// encoded targets, gt boxes in xyxy.
__global__ void k_prep(const float* __restrict__ gt, const float* __restrict__ anc,
                       int* __restrict__ bestbox, float* __restrict__ enc,
                       float* __restrict__ gtx, unsigned char* __restrict__ flag) {
    int i = blockIdx.x * blockDim.x + threadIdx.x;
    if (i >= NGT) return;
    int b = i / NN;
    const float* g = gt + (size_t)i * 7;
    float g0 = g[0], g1 = g[1], g2 = g[2], g3 = g[3], gw = g[4], gh = g[5];
    float garea = gw * gh;
    int bp = 0; float best = -1.0f;
    for (int a = 0; a < AA; ++a) {
        float aw = anc[a * 2], ah = anc[a * 2 + 1];
        float inter = fminf(gw, aw) * fminf(gh, ah);
        float v = inter / (garea + aw * ah - inter);
        if (v > best) { best = v; bp = a; }
    }
    int gxi = (int)g2, gyi = (int)g3;
    int bb = bp * (GG * GG) + gyi * GG + gxi;
    bestbox[i] = bb;
    flag[b * NBX + bb] = 1;               // benign races: all write 1
    enc[i * 4 + 0] = g0;
    enc[i * 4 + 1] = g1;
    enc[i * 4 + 2] = __logf(gw) - __logf(anc[bp * 2]);
    enc[i * 4 + 3] = __logf(gh) - __logf(anc[bp * 2 + 1]);
    float cx = g0 + g2 * (1.0f / GG), cy = g1 + g3 * (1.0f / GG);
    gtx[i * 4 + 0] = cx - gw * 0.5f;
    gtx[i * 4 + 1] = cy - gh * 0.5f;
    gtx[i * 4 + 2] = cx + gw * 0.5f;
    gtx[i * 4 + 3] = cy + gh * 0.5f;
}

// ---------------- kernel 2: all 327,680 boxes ----------------
// decode box, max IoU over this image's 50 GT boxes (cached in LDS),
// noobj + prior terms, deterministic LDS tree reduction.
__global__ void __launch_bounds__(256)
k_main(const float* __restrict__ pred, const float* __restrict__ anc,
       const float* __restrict__ gtx, const unsigned char* __restrict__ flag,
       const int* __restrict__ seenp, float* __restrict__ partials) {
    __shared__ float sg[NN * 4];
    __shared__ float sa[NN];
    __shared__ float sred[256];
    int b = blockIdx.y;
    int tid = threadIdx.x;
    if (tid < NN) {
        float x1 = gtx[(b * NN + tid) * 4 + 0];
        float y1 = gtx[(b * NN + tid) * 4 + 1];
        float x2 = gtx[(b * NN + tid) * 4 + 2];
        float y2 = gtx[(b * NN + tid) * 4 + 3];
        sg[tid * 4 + 0] = x1; sg[tid * 4 + 1] = y1;
        sg[tid * 4 + 2] = x2; sg[tid * 4 + 3] = y2;
        sa[tid] = (x2 - x1) * (y2 - y1);
    }
    __syncthreads();

    int box = blockIdx.x * 256 + tid;          // < NBX
    int a = box >> 10;
    int rem = box & 1023;
    int y = rem >> 5, x = rem & 31;
    size_t base = ((size_t)(b * (AA * CH) + a * CH)) * 1024 + rem;
    float p0 = pred[base];
    float p1 = pred[base + 1024];
    float p2 = pred[base + 2048];
    float p3 = pred[base + 3072];
    float p4 = pred[base + 4096];
    float tx = sigmoidf(p0), ty = sigmoidf(p1), tobj = sigmoidf(p4);
    float cx = tx + (float)x * (1.0f / GG);
    float cy = ty + (float)y * (1.0f / GG);
    float pw = anc[a * 2] * __expf(p2);
    float ph = anc[a * 2 + 1] * __expf(p3);
    float bx1 = cx - pw * 0.5f, by1 = cy - ph * 0.5f;
    float bx2 = cx + pw * 0.5f, by2 = cy + ph * 0.5f;
    float areab = pw * ph;

    float maxo = 0.0f;
    #pragma unroll 5
    for (int n = 0; n < NN; ++n) {
        float v = iou_xyxy(sg[n * 4 + 0], sg[n * 4 + 1], sg[n * 4 + 2], sg[n * 4 + 3],
                           sa[n], bx1, by1, bx2, by2, areab);
        maxo = fmaxf(maxo, v);
    }
    int posf = flag[b * NBX + box] ? 1 : ((maxo > IOU_THR) ? 2 : 0);

    float acc = 0.0f;
    if (posf == 0) acc += L_NOOBJ * tobj * tobj;                   // noobj term
    if (seenp[0] < SEEN_THR && posf != 1) {                        // prior term
        float d0 = tx - 0.5f / GG;
        float d1 = ty - 0.5f / GG;
        acc += L_PRIOR * (d0 * d0 + d1 * d1 + p2 * p2 + p3 * p3);
    }
    sred[tid] = acc;
    __syncthreads();
    for (int s = 128; s > 0; s >>= 1) {
        if (tid < s) sred[tid] += sred[tid + s];
        __syncthreads();
    }
    if (tid == 0) partials[blockIdx.y * gridDim.x + blockIdx.x] = sred[0];
}

// ---------------- kernel 3: per-GT terms (warp per GT, wave32) ----------------
// warp-cooperative softmax over the 80 class channels at the matched box,
// cls + 5*obj + coord.
__global__ void __launch_bounds__(256)
k_gt(const float* __restrict__ pred, const float* __restrict__ gt,
     const float* __restrict__ anc, const int* __restrict__ bestbox,
     const float* __restrict__ enc, const float* __restrict__ gtx,
     float* __restrict__ partials) {
    __shared__ float swarp[8];
    int tid  = threadIdx.x;
    int lane = tid & 31;
    int w    = tid >> 5;
    int gi   = blockIdx.x * 8 + w;             // < NGT exactly
    int b    = gi / NN;
    int bb   = bestbox[gi];
    int a    = bb >> 10;
    int rem  = bb & 1023;
    size_t cb = ((size_t)(b * (AA * CH) + a * CH)) * 1024 + rem;

    // classes: lane handles c = lane, lane+32, and lane+64 (lane<16)
    float l0 = pred[cb + (size_t)(5 + lane) * 1024];
    float l1 = pred[cb + (size_t)(5 + lane + 32) * 1024];
    float l2 = (lane < 16) ? pred[cb + (size_t)(5 + lane + 64) * 1024]
                           : -3.402823466e+38f;
    float m = fmaxf(fmaxf(l0, l1), l2);
    for (int off = 16; off > 0; off >>= 1) m = fmaxf(m, __shfl_xor(m, off));
    float e0 = __expf(l0 - m);
    float e1 = __expf(l1 - m);
    float e2 = (lane < 16) ? __expf(l2 - m) : 0.0f;
    float s = e0 + e1 + e2;
    for (int off = 16; off > 0; off >>= 1) s += __shfl_xor(s, off);
    float inv = 1.0f / s;
    int tcls = (int)gt[(size_t)gi * 7 + 6];
    float q0 = e0 * inv - ((lane == tcls) ? 1.0f : 0.0f);
    float q1 = e1 * inv - ((lane + 32 == tcls) ? 1.0f : 0.0f);
    float q2 = (lane < 16) ? (e2 * inv - ((lane + 64 == tcls) ? 1.0f : 0.0f)) : 0.0f;
    float csum = q0 * q0 + q1 * q1 + q2 * q2;
    for (int off = 16; off > 0; off >>= 1) csum += __shfl_xor(csum, off);

    float acc = 0.0f;
    if (lane == 0) {
        float p0 = pred[cb];
        float p1 = pred[cb + 1024];
        float p2 = pred[cb + 2048];
        float p3 = pred[cb + 3072];
        float p4 = pred[cb + 4096];
        float tx = sigmoidf(p0), ty = sigmoidf(p1), tobj = sigmoidf(p4);
        float d0 = tx - enc[gi * 4 + 0];
        float d1 = ty - enc[gi * 4 + 1];
        float d2 = p2 - enc[gi * 4 + 2];
        float d3 = p3 - enc[gi * 4 + 3];
        float coord = d0 * d0 + d1 * d1 + d2 * d2 + d3 * d3;
        int yy = rem >> 5, xx = rem & 31;
        float cx = tx + (float)xx * (1.0f / GG);
        float cy = ty + (float)yy * (1.0f / GG);
        float pw = anc[a * 2] * __expf(p2);
        float ph = anc[a * 2 + 1] * __expf(p3);
        float gx1 = gtx[gi * 4 + 0], gy1 = gtx[gi * 4 + 1];
        float gx2 = gtx[gi * 4 + 2], gy2 = gtx[gi * 4 + 3];
        float iou_t = iou_xyxy(gx1, gy1, gx2, gy2, (gx2 - gx1) * (gy2 - gy1),
                               cx - pw * 0.5f, cy - ph * 0.5f,
                               cx + pw * 0.5f, cy + ph * 0.5f, pw * ph);
        float dob = tobj - iou_t;
        acc = csum + L_OBJ * dob * dob + L_COORD * coord;
        swarp[w] = acc;
    }
    __syncthreads();
    if (tid == 0) {
        float t = 0.0f;
        for (int i = 0; i < 8; ++i) t += swarp[i];      // fixed order
        partials[MAINBLKS + blockIdx.x] = t;
    }
}

// ---------------- kernel 4: final deterministic reduction via WMMA ----------------
// Each lane holds a fixed-order strided partial sum s. A(16x4) = {s, 0} per lane,
// B(4x16) = all ones  =>  D[m][n] = s_m + s_{m+16} exactly (x1.0 mul is exact).
// Per ISA layout, lane0 VGPRs 0..7 hold D[0..7][0] and lane16 holds D[8..15][0];
// one shfl_xor(16) combine gives the exact wave sum on lane 0.
__global__ void __launch_bounds__(256)
k_final(const float* __restrict__ partials, int n, float* __restrict__ out) {
    __shared__ float sw[8];
    int tid = threadIdx.x;
    float s = 0.0f;
    for (int i = tid; i < n; i += 256) s += partials[i];   // fixed order

    v2f av; av.x = s;    av.y = 0.0f;
    v2f bv; bv.x = 1.0f; bv.y = 1.0f;
    v8f cv = {};
    v8f dv = __builtin_amdgcn_wmma_f32_16x16x4_f32(
        /*neg_a=*/false, av, /*neg_b=*/false, bv,
        /*c_mod=*/(short)0, cv, /*reuse_a=*/false, /*reuse_b=*/false);
    float t = dv[0] + dv[1] + dv[2] + dv[3] + dv[4] + dv[5] + dv[6] + dv[7];
    t += __shfl_xor(t, 16);                 // lane0: full wave32 sum

    int lane = tid & 31;
    if (lane == 0) sw[tid >> 5] = t;
    __syncthreads();
    if (tid == 0) {
        float tot = 0.0f;
        for (int i = 0; i < 8; ++i) tot += sw[i];          // fixed order
        out[0] = tot;
    }
}

extern "C" void kernel_launch(void* const* d_in, const int* in_sizes, int n_in,
                              void* d_out, int out_size, void* d_ws, size_t ws_size,
                              hipStream_t stream) {
    const float* pred = (const float*)d_in[0];   // (64, 425, 32, 32) f32
    const float* gtb  = (const float*)d_in[1];   // (64, 50, 7) f32
    const float* anc  = (const float*)d_in[2];   // (1, 5, 2, 1, 1) f32
    const int*   seen = (const int*)d_in[3];     // scalar

    char* ws = (char*)d_ws;
    int*           bestbox = (int*)(ws + WS_BESTBOX);
    float*         enc     = (float*)(ws + WS_ENC);
    float*         gtx     = (float*)(ws + WS_GTX);
    unsigned char* flag    = (unsigned char*)(ws + WS_FLAG);
    float*         part    = (float*)(ws + WS_PART);
    float*         out     = (float*)d_out;

    // 0) zero matched-box flags (ws is not re-poisoned between replays)
    k_zero_flags<<<(BB * NBX / 4 + 255) / 256, 256, 0, stream>>>((unsigned int*)flag);

    // 1) per-GT prep: anchor argmax, best_box scatter, targets, gt xyxy
    k_prep<<<(NGT + 255) / 256, 256, 0, stream>>>(gtb, anc, bestbox, enc, gtx, flag);

    // 2) all boxes: decode, 50-way IoU max, noobj + prior
    dim3 g2(NBX / 256, BB);
    k_main<<<g2, 256, 0, stream>>>(pred, anc, gtx, flag, seen, part);

    // 3) per-GT: softmax cls + obj + coord (warp per GT)
    k_gt<<<GTBLKS, 256, 0, stream>>>(pred, gtb, anc, bestbox, enc, gtx, part);

    // 4) deterministic final reduction (WMMA f32 16x16x4 wave-sum)
    k_final<<<1, 256, 0, stream>>>(part, NPART, out);
}